// LINKX_68143951118631
// MI455X (gfx1250) — compile-verified
//
#include <hip/hip_runtime.h>
#include <hip/hip_bf16.h>

// ---------------------------------------------------------------------------
// LINKX-style GNN layer for MI455X (gfx1250, wave32, WMMA).
//   agg  = D^-1/2 (A + I) D^-1/2 @ x          (atomic scatter-add)
//   out  = relu(concat(mlp_x(x), mlp_a(agg)) @ Wc + bc) @ Wcls + bcls
// All dense math uses V_WMMA_F32_16X16X4_F32 (exact f32, memory-bound anyway).
// ---------------------------------------------------------------------------

typedef __attribute__((ext_vector_type(2))) float v2f;
typedef __attribute__((ext_vector_type(8))) float v8f;

constexpr int IN_F  = 64;
constexpr int HID   = 128;
constexpr int OUT_F = 64;
constexpr int WAVES_PER_BLOCK = 4;   // 128 threads/block
constexpr int ROWS_PER_WAVE   = 16;  // one 16-row WMMA tile per wave

__device__ __forceinline__ v8f wmma4(v2f a, v2f b, v8f c) {
  // (neg_a, A, neg_b, B, c_mod, C, reuse_a, reuse_b)
  return __builtin_amdgcn_wmma_f32_16x16x4_f32(false, a, false, b, (short)0, c,
                                               false, false);
}

// A fragment (16x4 f32): lanes 0-15 hold M=0..15 {K=k0,k0+1}; lanes 16-31 {K=k0+2,k0+3}
__device__ __forceinline__ v2f a_frag_global(const float* __restrict__ A,
                                             int row_base, int lda, int k0,
                                             int lane, int nrows) {
  int m  = lane & 15;
  int kk = k0 + ((lane >> 4) << 1);
  int r  = row_base + m;
  if (r >= nrows) r = nrows - 1;  // clamp (stores are guarded separately)
  const float* p = A + (size_t)r * lda + kk;
  v2f v; v[0] = p[0]; v[1] = p[1];
  return v;
}

__device__ __forceinline__ v2f a_frag_lds(const float* s, int lda, int k0,
                                          int lane) {
  int m  = lane & 15;
  int kk = k0 + ((lane >> 4) << 1);
  const float* p = s + m * lda + kk;
  v2f v; v[0] = p[0]; v[1] = p[1];   // -> ds_load_b64
  return v;
}

// B fragment (4x16 f32): lane n=lane&15; lanes 0-15 rows {k0,k0+1}, lanes 16-31 {k0+2,k0+3}
__device__ __forceinline__ v2f b_frag_global(const float* __restrict__ B,
                                             int ldb, int k0, int n0, int lane) {
  int n  = n0 + (lane & 15);
  int kk = k0 + ((lane >> 4) << 1);
  const float* p = B + (size_t)kk * ldb + n;
  v2f v; v[0] = p[0]; v[1] = p[ldb];
  return v;
}

// D tile (16x16 f32): VGPR v, lane l -> row (v + 8*(l>>4)), col (l&15). Stage to LDS row-major.
__device__ __forceinline__ void d_to_lds(float* s, int ldd, int n0, int lane,
                                         v8f d) {
  int half = lane >> 4;
  int nn   = n0 + (lane & 15);
#pragma unroll
  for (int v = 0; v < 8; ++v) s[(v + 8 * half) * ldd + nn] = d[v];
}

// ---------------------------------------------------------------------------
// Aggregation kernels
// ---------------------------------------------------------------------------
__global__ void k_deg_init(float* __restrict__ deg, int n) {
  int i = blockIdx.x * blockDim.x + threadIdx.x;
  if (i < n) deg[i] = 1.0f;  // self loop
}

__global__ void k_deg_count(const long long* __restrict__ ei,
                            float* __restrict__ deg, int e) {
  int i = blockIdx.x * blockDim.x + threadIdx.x;
  if (i < e) atomicAdd(&deg[ei[i]], 1.0f);  // rows are ei[0:E]
}

__global__ void k_dinv(float* __restrict__ deg, int n) {
  int i = blockIdx.x * blockDim.x + threadIdx.x;
  if (i < n) {
    float d = deg[i];
    deg[i] = d > 0.0f ? rsqrtf(d) : 0.0f;  // in-place: deg -> dinv
  }
}

__global__ void k_agg_init(const float* __restrict__ x,
                           const float* __restrict__ dinv,
                           float* __restrict__ agg, int n64) {
  int i = blockIdx.x * blockDim.x + threadIdx.x;
  if (i < n64) {
    float di = dinv[i >> 6];
    agg[i] = x[i] * di * di;  // self-loop contribution
  }
}

// 16 lanes per edge, float4 per lane -> global_load_b128 + 4x global_atomic_add_f32
__global__ void k_scatter(const long long* __restrict__ ei,
                          const float* __restrict__ x,
                          const float* __restrict__ dinv,
                          float* __restrict__ agg, int E) {
  long long t = (long long)blockIdx.x * blockDim.x + threadIdx.x;
  long long e = t >> 4;
  if (e >= E) return;
  int c = ((int)t & 15) << 2;
  long long r  = ei[e];
  long long ci = ei[(long long)E + e];
  float w = dinv[r] * dinv[ci];
  const float4 xv = *reinterpret_cast<const float4*>(x + ci * IN_F + c);
  float* dst = agg + r * IN_F + c;
  atomicAdd(dst + 0, xv.x * w);
  atomicAdd(dst + 1, xv.y * w);
  atomicAdd(dst + 2, xv.z * w);
  atomicAdd(dst + 3, xv.w * w);
}

// ---------------------------------------------------------------------------
// Fused 6-GEMM MLP: each wave handles 16 node rows end-to-end.
// Per-wave private LDS: buf1/buf2 = 16x128 f32 each (no cross-wave sync needed).
// ---------------------------------------------------------------------------
__global__ void __launch_bounds__(WAVES_PER_BLOCK * 32)
k_mlp_fused(const float* __restrict__ x, const float* __restrict__ agg,
            const float* __restrict__ W1x, const float* __restrict__ b1x,
            const float* __restrict__ W2x, const float* __restrict__ b2x,
            const float* __restrict__ W1a, const float* __restrict__ b1a,
            const float* __restrict__ W2a, const float* __restrict__ b2a,
            const float* __restrict__ Wc,  const float* __restrict__ bc,
            const float* __restrict__ Wcls,const float* __restrict__ bcls,
            float* __restrict__ out, int nnodes) {
  extern __shared__ float smem[];
  const int lane = threadIdx.x & 31;
  const int wave = threadIdx.x >> 5;
  const int rowbase = (blockIdx.x * WAVES_PER_BLOCK + wave) * ROWS_PER_WAVE;
  if (rowbase >= nnodes) return;  // wave-uniform exit keeps EXEC full for WMMA

  float* buf1 = smem + wave * (ROWS_PER_WAVE * HID);
  float* buf2 = smem + (WAVES_PER_BLOCK + wave) * (ROWS_PER_WAVE * HID);

  // ---- stage 1: h1x = relu(x @ W1x + b1x) -> buf1
  for (int nt = 0; nt < HID / 16; ++nt) {
    v8f acc = {};
#pragma unroll
    for (int k0 = 0; k0 < IN_F; k0 += 4)
      acc = wmma4(a_frag_global(x, rowbase, IN_F, k0, lane, nnodes),
                  b_frag_global(W1x, HID, k0, nt * 16, lane), acc);
    float bv = b1x[nt * 16 + (lane & 15)];
#pragma unroll
    for (int v = 0; v < 8; ++v) acc[v] = fmaxf(acc[v] + bv, 0.0f);
    d_to_lds(buf1, HID, nt * 16, lane, acc);
  }

  // ---- stage 2: hx = h1x @ W2x + b2x -> buf2
  for (int nt = 0; nt < HID / 16; ++nt) {
    v8f acc = {};
#pragma unroll
    for (int k0 = 0; k0 < HID; k0 += 4)
      acc = wmma4(a_frag_lds(buf1, HID, k0, lane),
                  b_frag_global(W2x, HID, k0, nt * 16, lane), acc);
    float bv = b2x[nt * 16 + (lane & 15)];
#pragma unroll
    for (int v = 0; v < 8; ++v) acc[v] += bv;
    d_to_lds(buf2, HID, nt * 16, lane, acc);
  }

  // ---- stage 3: h1a = relu(agg @ W1a + b1a) -> buf1 (reuse)
  for (int nt = 0; nt < HID / 16; ++nt) {
    v8f acc = {};
#pragma unroll
    for (int k0 = 0; k0 < IN_F; k0 += 4)
      acc = wmma4(a_frag_global(agg, rowbase, IN_F, k0, lane, nnodes),
                  b_frag_global(W1a, HID, k0, nt * 16, lane), acc);
    float bv = b1a[nt * 16 + (lane & 15)];
#pragma unroll
    for (int v = 0; v < 8; ++v) acc[v] = fmaxf(acc[v] + bv, 0.0f);
    d_to_lds(buf1, HID, nt * 16, lane, acc);
  }

  // ---- stage 4: ha = h1a @ W2a + b2a (regs), then ha -> buf1
  v8f ha[HID / 16];
  for (int nt = 0; nt < HID / 16; ++nt) {
    v8f acc = {};
#pragma unroll
    for (int k0 = 0; k0 < HID; k0 += 4)
      acc = wmma4(a_frag_lds(buf1, HID, k0, lane),
                  b_frag_global(W2a, HID, k0, nt * 16, lane), acc);
    float bv = b2a[nt * 16 + (lane & 15)];
#pragma unroll
    for (int v = 0; v < 8; ++v) acc[v] += bv;
    ha[nt] = acc;
  }
  for (int nt = 0; nt < HID / 16; ++nt) d_to_lds(buf1, HID, nt * 16, lane, ha[nt]);

  // ---- stage 5: h = relu(hx @ Wc[0:128] + ha @ Wc[128:256] + bc) (regs), then -> buf1
  v8f hh[HID / 16];
  for (int nt = 0; nt < HID / 16; ++nt) {
    v8f acc = {};
#pragma unroll
    for (int k0 = 0; k0 < HID; k0 += 4)
      acc = wmma4(a_frag_lds(buf2, HID, k0, lane),
                  b_frag_global(Wc, HID, k0, nt * 16, lane), acc);
#pragma unroll
    for (int k0 = 0; k0 < HID; k0 += 4)
      acc = wmma4(a_frag_lds(buf1, HID, k0, lane),
                  b_frag_global(Wc + (size_t)HID * HID, HID, k0, nt * 16, lane),
                  acc);
    float bv = bc[nt * 16 + (lane & 15)];
#pragma unroll
    for (int v = 0; v < 8; ++v) acc[v] = fmaxf(acc[v] + bv, 0.0f);
    hh[nt] = acc;
  }
  for (int nt = 0; nt < HID / 16; ++nt) d_to_lds(buf1, HID, nt * 16, lane, hh[nt]);

  // ---- stage 6: out = h @ Wcls + bcls -> global (D layout scatter, row-guarded)
  for (int nt = 0; nt < OUT_F / 16; ++nt) {
    v8f acc = {};
#pragma unroll
    for (int k0 = 0; k0 < HID; k0 += 4)
      acc = wmma4(a_frag_lds(buf1, HID, k0, lane),
                  b_frag_global(Wcls, OUT_F, k0, nt * 16, lane), acc);
    int nn  = nt * 16 + (lane & 15);
    float bv = bcls[nn];
    int half = lane >> 4;
#pragma unroll
    for (int v = 0; v < 8; ++v) {
      int r = rowbase + v + 8 * half;
      if (r < nnodes) out[(size_t)r * OUT_F + nn] = acc[v] + bv;
    }
  }
}

// ---------------------------------------------------------------------------
// Host launcher
// ---------------------------------------------------------------------------
extern "C" void kernel_launch(void* const* d_in, const int* in_sizes, int n_in,
                              void* d_out, int out_size, void* d_ws, size_t ws_size,
                              hipStream_t stream) {
  const float*     x    = (const float*)d_in[0];
  const long long* ei   = (const long long*)d_in[1];   // int64 (2, E)
  const float* W1x  = (const float*)d_in[2];
  const float* b1x  = (const float*)d_in[3];
  const float* W2x  = (const float*)d_in[4];
  const float* b2x  = (const float*)d_in[5];
  const float* W1a  = (const float*)d_in[6];
  const float* b1a  = (const float*)d_in[7];
  const float* W2a  = (const float*)d_in[8];
  const float* b2a  = (const float*)d_in[9];
  const float* Wc   = (const float*)d_in[10];
  const float* bc   = (const float*)d_in[11];
  const float* Wcls = (const float*)d_in[12];
  const float* bcls = (const float*)d_in[13];
  float* out = (float*)d_out;

  const int N = in_sizes[0] / IN_F;
  const int E = in_sizes[1] / 2;

  // Workspace: agg (N*64 f32) then deg/dinv (N f32)
  float* agg = (float*)d_ws;
  float* deg = (float*)((char*)d_ws + (size_t)N * IN_F * sizeof(float));

  const int T = 256;
  k_deg_init <<<(N + T - 1) / T, T, 0, stream>>>(deg, N);
  k_deg_count<<<(E + T - 1) / T, T, 0, stream>>>(ei, deg, E);
  k_dinv     <<<(N + T - 1) / T, T, 0, stream>>>(deg, N);
  k_agg_init <<<((N * IN_F) + T - 1) / T, T, 0, stream>>>(x, deg, agg, N * IN_F);
  long long st = (long long)E * 16;
  k_scatter  <<<(int)((st + T - 1) / T), T, 0, stream>>>(ei, x, deg, agg, E);

  const int rows_per_block = WAVES_PER_BLOCK * ROWS_PER_WAVE;  // 64
  const int grid = (N + rows_per_block - 1) / rows_per_block;
  const size_t shmem = 2u * WAVES_PER_BLOCK * ROWS_PER_WAVE * HID * sizeof(float); // 64 KB
  k_mlp_fused<<<grid, WAVES_PER_BLOCK * 32, shmem, stream>>>(
      x, agg, W1x, b1x, W2x, b2x, W1a, b1a, W2a, b2a, Wc, bc, Wcls, bcls, out, N);
}